// SOM_50852412785301
// MI455X (gfx1250) — compile-verified
//
#include <hip/hip_runtime.h>

typedef __attribute__((ext_vector_type(2))) float v2f;
typedef __attribute__((ext_vector_type(8))) float v8f;

#define XDIM 256
#define YDIM 256
#define NPART (XDIM * YDIM)      // 65536
#define DIMV 512
#define NUM_ITERS_F 100.0f
#define LEARNING_RADIUS 0.3f
#define COGNITIVE 0.1f
#define SOCIAL 0.1f
#define INERTIA 0.5f
#define SIGMA 128.0f             // max(X,Y)/2

// ---------------------------------------------------------------------------
// Kernel 1: per-block BMU partial. Each wave computes 16 squared distances
// via V_WMMA_F32_16X16X4_F32: acc += diff * diff^T (Gram); diagonal = ||diff||^2.
// A (16x4) and B (4x16 = A^T) have identical lane/VGPR layouts, so the same
// register pair feeds both operands.
// ---------------------------------------------------------------------------
__global__ __launch_bounds__(256) void som_bmu_partial(
    const float* __restrict__ input,
    const float* __restrict__ particles,
    float* __restrict__ wsF, int* __restrict__ wsI)
{
    __shared__ float s_in[DIMV];
    __shared__ float s_dist[128];
    __shared__ float s_rd[128];
    __shared__ int   s_ri[128];

    const int tid = threadIdx.x;
    for (int i = tid; i < DIMV; i += 256) s_in[i] = input[i];
    __syncthreads();

    const int wave  = tid >> 5;
    const int lane  = tid & 31;
    const int row   = lane & 15;          // M index 0..15 (both lane halves)
    const int khalf = (lane >> 4) << 1;   // 0 for lanes 0-15, 2 for lanes 16-31
    const int blockStart = blockIdx.x * 128;
    const int pbase = blockStart + wave * 16;

    const float* prow = particles + (size_t)(pbase + row) * DIMV;

    v8f acc = {};
    #pragma unroll 4
    for (int kb = 0; kb < DIMV; kb += 4) {
        const int kc = kb + khalf;
        v2f p2  = *(const v2f*)(prow + kc);    // global_load_b64
        v2f in2 = *(const v2f*)(s_in + kc);    // ds_load_b64
        v2f d;
        d.x = in2.x - p2.x;
        d.y = in2.y - p2.y;
        // D = A*B + C, A = B = diff tile -> Gram matrix, fp32 accumulate
        acc = __builtin_amdgcn_wmma_f32_16x16x4_f32(
            false, d, false, d, (short)0, acc, false, false);
    }

    // Diagonal extraction: element (M=v,N=v) lives in VGPR v at lane v (v<8);
    // element (M=v+8,N=v+8) lives in VGPR v at lane v+24.
    #pragma unroll
    for (int v = 0; v < 8; ++v) {
        if (lane == v)      s_dist[wave * 16 + v]     = acc[v];
        if (lane == v + 24) s_dist[wave * 16 + v + 8] = acc[v];
    }
    __syncthreads();

    if (tid < 128) {
        s_rd[tid] = s_dist[tid];
        s_ri[tid] = blockStart + tid;
    }
    __syncthreads();
    for (int s = 64; s > 0; s >>= 1) {
        if (tid < s) {
            float d1 = s_rd[tid], d2 = s_rd[tid + s];
            int   i1 = s_ri[tid], i2 = s_ri[tid + s];
            if (d2 < d1 || (d2 == d1 && i2 < i1)) { s_rd[tid] = d2; s_ri[tid] = i2; }
        }
        __syncthreads();
    }
    if (tid == 0) { wsF[blockIdx.x] = s_rd[0]; wsI[blockIdx.x] = s_ri[0]; }
}

// ---------------------------------------------------------------------------
// Kernel 2: reduce the 512 block partials -> bmu_idx; precompute mask params.
// ---------------------------------------------------------------------------
__global__ __launch_bounds__(256) void som_bmu_final(
    const float* __restrict__ wsF, const int* __restrict__ wsI,
    const int* __restrict__ grid_locations,
    const int* __restrict__ iter_num,
    int* __restrict__ wsR, float* __restrict__ wsP)
{
    __shared__ float s_rd[256];
    __shared__ int   s_ri[256];
    const int tid = threadIdx.x;

    float d1 = wsF[tid];       int i1 = wsI[tid];
    float d2 = wsF[tid + 256]; int i2 = wsI[tid + 256];
    if (d2 < d1 || (d2 == d1 && i2 < i1)) { d1 = d2; i1 = i2; }
    s_rd[tid] = d1; s_ri[tid] = i1;
    __syncthreads();
    for (int s = 128; s > 0; s >>= 1) {
        if (tid < s) {
            float a = s_rd[tid], b = s_rd[tid + s];
            int   ia = s_ri[tid], ib = s_ri[tid + s];
            if (b < a || (b == a && ib < ia)) { s_rd[tid] = b; s_ri[tid] = ib; }
        }
        __syncthreads();
    }
    if (tid == 0) {
        const int bmu = s_ri[0];
        wsR[0] = bmu;
        const float bx = (float)grid_locations[2 * bmu];
        const float by = (float)grid_locations[2 * bmu + 1];
        const float decay  = 1.0f - (float)iter_num[0] / NUM_ITERS_F;
        const float radius = LEARNING_RADIUS * decay;
        const float sd     = SIGMA * decay;
        wsP[0] = bx;
        wsP[1] = by;
        wsP[2] = radius;
        wsP[3] = 1.0f / (sd * sd);
    }
}

// ---------------------------------------------------------------------------
// Kernel 3: streaming PSO update, float4 per thread (global_load/store_b128).
// out[0 .. N*DIM)      = new_particles
// out[N*DIM .. 2N*DIM) = new_velocities
// ---------------------------------------------------------------------------
__global__ __launch_bounds__(256) void som_update(
    const float* __restrict__ particles,
    const float* __restrict__ velocities,
    const float* __restrict__ r1,
    const float* __restrict__ r2,
    const int*   __restrict__ grid_locations,
    const int*   __restrict__ wsR, const float* __restrict__ wsP,
    float* __restrict__ out)
{
    const size_t id  = (size_t)blockIdx.x * blockDim.x + threadIdx.x; // float4 idx
    const int    row = (int)(id >> 7);          // 128 float4 per row
    const int    j   = (int)(id & 127) << 2;

    const int   bmu    = wsR[0];
    const float bx     = wsP[0];
    const float by     = wsP[1];
    const float radius = wsP[2];
    const float inv_s2 = wsP[3];

    const float gx = (float)grid_locations[2 * row]     - bx;
    const float gy = (float)grid_locations[2 * row + 1] - by;
    const float d2 = gx * gx + gy * gy;
    // global_nbest = exp(0) = 1; mask: 1 - neighborhood <= radius
    const bool upd = (1.0f - expf(-d2 * inv_s2)) <= radius;

    const size_t off = (size_t)row * DIMV + j;
    const float4 p  = *(const float4*)(particles  + off);
    const float4 v  = *(const float4*)(velocities + off);
    const float4 a1 = *(const float4*)(r1 + off);
    const float4 a2 = *(const float4*)(r2 + off);
    const float4 gb = *(const float4*)(particles + (size_t)bmu * DIMV + j);

    float4 df, vn, np, nv;
    df.x = gb.x - p.x; df.y = gb.y - p.y; df.z = gb.z - p.z; df.w = gb.w - p.w;
    vn.x = INERTIA * v.x + COGNITIVE * a1.x * df.x + SOCIAL * a2.x * df.x;
    vn.y = INERTIA * v.y + COGNITIVE * a1.y * df.y + SOCIAL * a2.y * df.y;
    vn.z = INERTIA * v.z + COGNITIVE * a1.z * df.z + SOCIAL * a2.z * df.z;
    vn.w = INERTIA * v.w + COGNITIVE * a1.w * df.w + SOCIAL * a2.w * df.w;

    if (upd) {
        np.x = p.x + vn.x; np.y = p.y + vn.y; np.z = p.z + vn.z; np.w = p.w + vn.w;
        nv = vn;
    } else {
        np = p;
        nv = v;
    }

    *(float4*)(out + off) = np;
    *(float4*)(out + (size_t)NPART * DIMV + off) = nv;
}

// ---------------------------------------------------------------------------
extern "C" void kernel_launch(void* const* d_in, const int* in_sizes, int n_in,
                              void* d_out, int out_size, void* d_ws, size_t ws_size,
                              hipStream_t stream) {
    const float* input      = (const float*)d_in[0];
    const float* particles  = (const float*)d_in[1];
    const float* velocities = (const float*)d_in[2];
    const float* r1         = (const float*)d_in[3];
    const float* r2         = (const float*)d_in[4];
    const int*   grid_loc   = (const int*)d_in[5];
    const int*   iter_num   = (const int*)d_in[6];
    float*       out        = (float*)d_out;

    float* wsF = (float*)d_ws;              // 512 partial dists
    int*   wsI = (int*)d_ws + 512;          // 512 partial indices
    int*   wsR = (int*)d_ws + 1024;         // bmu_idx
    float* wsP = (float*)d_ws + 1028;       // bx, by, radius, 1/sigma_d^2

    som_bmu_partial<<<NPART / 128, 256, 0, stream>>>(input, particles, wsF, wsI);
    som_bmu_final<<<1, 256, 0, stream>>>(wsF, wsI, grid_loc, iter_num, wsR, wsP);

    const int n4 = NPART * (DIMV / 4);      // 8,388,608 float4 elements
    som_update<<<n4 / 256, 256, 0, stream>>>(particles, velocities, r1, r2,
                                             grid_loc, wsR, wsP, out);
}